// MinimalLoss_1065151889702
// MI455X (gfx1250) — compile-verified
//
#include <hip/hip_runtime.h>
#include <hip/hip_bf16.h>
#include <stdint.h>

#define NCLS 80
#define NCH  85   // 5 + NCLS
#define TPB  256

typedef __attribute__((ext_vector_type(2))) float v2f;
typedef __attribute__((ext_vector_type(8))) float v8f;

__device__ __forceinline__ float sigmoidf_(float x) {
    return 1.0f / (1.0f + expf(-x));
}

// BCE term pair with -100 log clamp, matching the reference numerics.
__device__ __forceinline__ void bce_terms(float x, float& neg_lp, float& neg_lq) {
    float p  = sigmoidf_(x);
    neg_lp = -fmaxf(logf(p),        -100.0f);   // -clipped log(p)
    neg_lq = -fmaxf(logf(1.0f - p), -100.0f);   // -clipped log(1-p)
}

// Layout-independent wave32 sum using V_WMMA_F32_16X16X4_F32 with A == ones:
// D[m][n] = sum_k B[k][n]  =>  sum over row 0 of D == sum of ALL B entries.
// Each lane feeds its value into B.x (B.y = 0); 32 lanes fill the 4x16 B exactly.
// Caller: all 32 lanes of the wave must execute this (EXEC all ones), then
// lane 0 sums lds[0..15] (D VGPR0 over lanes 0-15 == row M=0).
__device__ __forceinline__ float wmma_wave_reduce_stage(float lane_val) {
    v2f a; a.x = 1.0f; a.y = 1.0f;         // A all-ones regardless of layout
    v2f b; b.x = lane_val; b.y = 0.0f;
    v8f c = {};
    v8f d = __builtin_amdgcn_wmma_f32_16x16x4_f32(
        /*neg_a=*/false, a, /*neg_b=*/false, b,
        /*c_mod=*/(short)0, c, /*reuse_a=*/false, /*reuse_b=*/false);
    return d[0];                            // lanes 0-15 hold D[0][n]
}

// ---------------------------------------------------------------------------
// Kernel 0: zero the obj-mask bytes (dword stores)
// ---------------------------------------------------------------------------
__global__ void k_zero_mask(uint32_t* mask32, int n32) {
    int i = blockIdx.x * blockDim.x + threadIdx.x;
    if (i < n32) mask32[i] = 0u;
}

// ---------------------------------------------------------------------------
// Kernel 1: per-target pass. Scatter obj mask, accumulate
// (sum_vf, sum_vf*per_xy, sum_vf*per_wh, sum_vf*per_cls) per block.
// ---------------------------------------------------------------------------
__global__ void k_targets(const float* __restrict__ pred,
                          const float* __restrict__ tgt,
                          const int* __restrict__ Hp,
                          const int* __restrict__ Wp,
                          unsigned char* __restrict__ mask,
                          float* __restrict__ tpart,
                          int cells, int totalT)
{
    const int tid = threadIdx.x;
    const int g   = blockIdx.x * blockDim.x + tid;
    const int H   = *Hp;
    const int W   = *Wp;
    const int HW  = H * W;
    const int B   = cells / HW;
    const int T   = totalT / B;

    float vf = 0.0f, sxy = 0.0f, swh = 0.0f, scls = 0.0f;

    if (g < totalT) {
        const float* t5 = tgt + (long)g * 5;
        float cls_id = t5[0], cx = t5[1], cy = t5[2], w = t5[3], h = t5[4];
        float fx = cx * (float)W;
        float fy = cy * (float)H;
        float gx = floorf(fx);
        float gy = floorf(fy);
        bool finite = isfinite(cls_id) && isfinite(cx) && isfinite(cy) &&
                      isfinite(w) && isfinite(h);
        bool valid = finite && (gx >= 0.0f) && (gx < (float)W) &&
                               (gy >= 0.0f) && (gy < (float)H);
        int gxi = min(max((int)gx, 0), W - 1);
        int gyi = min(max((int)gy, 0), H - 1);
        int b   = g / T;
        long cell = (long)b * HW + (long)(gyi * W + gxi);
        const float* p = pred + cell * NCH;

        // xy: mse over 2 elems of (sigmoid(p0,p1) - (cx*W-gx, cy*H-gy))
        float px = sigmoidf_(p[0]);
        float py = sigmoidf_(p[1]);
        float tx = fx - gx;
        float ty = fy - gy;
        float per_xy = 0.5f * ((px - tx) * (px - tx) + (py - ty) * (py - ty));

        // wh: mse over 2 elems of (exp(p2,p3) - (w*W, h*H))
        float pw = expf(p[2]);
        float ph = expf(p[3]);
        float tw = w * (float)W;
        float th = h * (float)H;
        float per_wh = 0.5f * ((pw - tw) * (pw - tw) + (ph - th) * (ph - th));

        // cls: mean BCE over NCLS with one-hot target
        int ci = (int)cls_id;
        float acc = 0.0f;
        #pragma unroll 4
        for (int c = 0; c < NCLS; ++c) {
            float nlp, nlq;
            bce_terms(p[5 + c], nlp, nlq);
            acc += (c == ci) ? nlp : nlq;
        }
        float per_cls = acc * (1.0f / (float)NCLS);

        if (valid) {
            vf = 1.0f;
            mask[cell] = 1;            // races write identical value: benign
        }
        sxy  = vf * per_xy;
        swh  = vf * per_wh;
        scls = vf * per_cls;
    }

    // Deterministic fixed-tree block reduction of the 4 sums.
    __shared__ float s0[TPB], s1[TPB], s2[TPB], s3[TPB];
    s0[tid] = vf; s1[tid] = sxy; s2[tid] = swh; s3[tid] = scls;
    __syncthreads();
    for (int s = TPB / 2; s > 0; s >>= 1) {
        if (tid < s) {
            s0[tid] += s0[tid + s];
            s1[tid] += s1[tid + s];
            s2[tid] += s2[tid + s];
            s3[tid] += s3[tid + s];
        }
        __syncthreads();
    }
    if (tid == 0) {
        float* o = tpart + (long)blockIdx.x * 4;
        o[0] = s0[0]; o[1] = s1[0]; o[2] = s2[0]; o[3] = s3[0];
    }
}

// ---------------------------------------------------------------------------
// Kernel 2: confidence BCE over all cells (HBM-bound, ~105 MB of lines).
// Per-block partial sum; wave0 finishes the block with a WMMA ones-reduce.
// ---------------------------------------------------------------------------
__global__ void k_conf(const float* __restrict__ pred,
                       const unsigned char* __restrict__ mask,
                       float* __restrict__ cpart,
                       int cells)
{
    const int tid = threadIdx.x;
    const int i   = blockIdx.x * TPB + tid;

    float bce = 0.0f;
    if (i < cells) {
        float x = pred[(long)i * NCH + 4];
        float nlp, nlq;
        bce_terms(x, nlp, nlq);
        bce = mask[i] ? nlp : nlq;   // obj mask is exactly 0 or 1
    }

    __shared__ float sm[TPB];
    sm[tid] = bce;
    __syncthreads();

    if (tid < 32) {                  // wave 0 entirely: EXEC all-ones
        float acc = 0.0f;
        #pragma unroll
        for (int k = 0; k < TPB / 32; ++k) acc += sm[tid + 32 * k];
        sm[tid] = wmma_wave_reduce_stage(acc);   // v_wmma_f32_16x16x4_f32
    }
    __syncthreads();
    if (tid == 0) {
        float s = 0.0f;
        #pragma unroll
        for (int n = 0; n < 16; ++n) s += sm[n];  // row 0 of D
        cpart[blockIdx.x] = s;
    }
}

// ---------------------------------------------------------------------------
// Kernel 3: final combine (1 block, 32 threads = one wave).
// ---------------------------------------------------------------------------
__global__ void k_final(const float* __restrict__ tpart, int nTB,
                        const float* __restrict__ cpart, int nCB,
                        float* __restrict__ out, int cells)
{
    const int lane = threadIdx.x;    // 32 lanes
    float acc = 0.0f;
    for (int j = lane; j < nCB; j += 32) acc += cpart[j];  // fixed per-lane order

    __shared__ float sm[32];
    sm[lane] = wmma_wave_reduce_stage(acc);   // collapse 32 lane partials
    __syncthreads();

    if (lane == 0) {
        float conf_sum = 0.0f;
        #pragma unroll
        for (int n = 0; n < 16; ++n) conf_sum += sm[n];

        float svf = 0.0f, sxy = 0.0f, swh = 0.0f, scls = 0.0f;
        for (int b = 0; b < nTB; ++b) {
            svf  += tpart[4 * b + 0];
            sxy  += tpart[4 * b + 1];
            swh  += tpart[4 * b + 2];
            scls += tpart[4 * b + 3];
        }
        float denom    = fmaxf(svf, 1.0f);
        float loss_xy  = sxy  / denom;
        float loss_wh  = swh  / denom;
        float loss_cls = scls / denom;
        float loss_conf = conf_sum / (float)cells;

        out[0] = loss_xy * 5.0f + loss_wh * 5.0f + loss_conf + loss_cls;
        out[1] = loss_xy;
        out[2] = loss_wh;
        out[3] = loss_conf;
        out[4] = loss_cls;
    }
}

// ---------------------------------------------------------------------------
extern "C" void kernel_launch(void* const* d_in, const int* in_sizes, int n_in,
                              void* d_out, int out_size, void* d_ws, size_t ws_size,
                              hipStream_t stream) {
    const float* pred = (const float*)d_in[0];   // [B, H*W, 85] f32
    const float* tgt  = (const float*)d_in[1];   // [B, T, 5]   f32
    const int*   Hp   = (const int*)d_in[2];     // scalar (read on device)
    const int*   Wp   = (const int*)d_in[3];     // scalar (read on device)
    float*       out  = (float*)d_out;           // 5 scalars

    const int cells  = in_sizes[0] / NCH;        // B * H * W
    const int totalT = in_sizes[1] / 5;          // B * T

    const int nTB = (totalT + TPB - 1) / TPB;    // target blocks
    const int nCB = (cells  + TPB - 1) / TPB;    // conf blocks
    const int n32 = (cells + 3) / 4;             // mask dwords

    // Workspace layout (poisoned by harness; fully re-initialized here):
    //   [0, n32*4)             : obj mask bytes
    //   [off1, off1+nTB*16)    : target partials (4 floats per block)
    //   then                   : conf partials  (1 float per block)
    unsigned char* mask = (unsigned char*)d_ws;
    size_t off1 = (((size_t)n32 * 4) + 255) & ~(size_t)255;
    float* tpart = (float*)((char*)d_ws + off1);
    float* cpart = tpart + (size_t)nTB * 4;

    k_zero_mask<<<(n32 + TPB - 1) / TPB, TPB, 0, stream>>>((uint32_t*)mask, n32);
    k_targets  <<<nTB, TPB, 0, stream>>>(pred, tgt, Hp, Wp, mask, tpart, cells, totalT);
    k_conf     <<<nCB, TPB, 0, stream>>>(pred, mask, cpart, cells);
    k_final    <<<1, 32, 0, stream>>>(tpart, nTB, cpart, nCB, out, cells);
}